// CRFLayer_66666482368754
// MI455X (gfx1250) — compile-verified
//
#include <hip/hip_runtime.h>

typedef float v2f __attribute__((ext_vector_type(2)));

static constexpr int NB = 2048;
static constexpr int NT = 512;
static constexpr int NK = 64;

// Uniform lane broadcast via v_readlane (wave32): result lands in an SGPR,
// consumed as the scalar operand of the packed add.
__device__ __forceinline__ float lane_bcast(float v, int i) {
  return __int_as_float(__builtin_amdgcn_readlane(__float_as_int(v), i));
}

__global__ __launch_bounds__(32, 1) void viterbi_fwd_bwd(
    const float* __restrict__ pot,     // [B, T, K]
    const float* __restrict__ trans,   // [K, K]
    const int*   __restrict__ slen,    // [B, 1]
    float*       __restrict__ out)     // [B*T] tags (as float) then [B] scores
{
  // 32 KB of backpointers. The first 16 KB double as the async staging buffer
  // for the transition matrix (fully consumed into VGPRs before bp row 1 is
  // ever written, and bp rows are write-before-read).
  __shared__ __align__(16) unsigned char bp[NT * NK];
  __shared__ unsigned int tagw[NT / 4];        // packed decoded tags (512 B)

  const int b = blockIdx.x;
  const int lane = threadIdx.x;                // 0..31, wave32
  const float* __restrict__ pb = pot + (size_t)b * NT * NK;
  const int L = slen[b];                       // 1..T

  // ---- Stage transitions (16 KB) into LDS with the async data mover ----
  {
    unsigned int laddr = (unsigned int)(uintptr_t)&bp[lane * 16];
    const void* gaddr = (const unsigned char*)trans + lane * 16;
#pragma unroll
    for (int k = 0; k < 32; ++k) {             // 32 lanes x 16 B = 512 B/op
      asm volatile("global_load_async_to_lds_b128 %0, %1, off offset:%2"
                   :: "v"(laddr), "v"(gaddr), "i"(k * 512) : "memory");
    }
    asm volatile("s_wait_asynccnt 0" ::: "memory");
  }
  __syncthreads();

  // Transition columns j and j+32 resident in VGPRs as float2 pairs
  // (conflict-free LDS reads: lane j always addresses bank j).
  v2f tc[NK];
  {
    const float* ldsT = (const float*)&bp[0];
#pragma unroll
    for (int i = 0; i < NK; ++i) {
      v2f t2;
      t2.x = ldsT[i * NK + lane];
      t2.y = ldsT[i * NK + lane + 32];
      tc[i] = t2;
    }
  }

  // alpha = potentials[b, 0, :]; lane j holds alpha[j], alpha[j+32].
  float a0 = pb[lane];
  float a1 = pb[lane + 32];
  // Emission double buffer (row t consumed while row t+1 is in flight).
  float e0 = pb[NK + lane];
  float e1 = pb[NK + lane + 32];

  // Forward DP only while t < seq_len: alphas freeze afterwards and the
  // backward pass never reads bp rows >= seq_len, so that work is dead.
#pragma unroll 1
  for (int t = 1; t < L; ++t) {
    int tpre = (t + 8 < NT) ? (t + 8) : (NT - 1);
    __builtin_prefetch(pb + (size_t)tpre * NK + lane, 0, 0);
    int tn = (t + 1 < NT) ? (t + 1) : (NT - 1);
    float ne0 = pb[(size_t)tn * NK + lane];
    float ne1 = pb[(size_t)tn * NK + lane + 32];

    // i = 0 initializes running max/argmax.
    float s0 = lane_bcast(a0, 0);
    v2f sv0; sv0.x = s0; sv0.y = s0;
    v2f c0 = sv0 + tc[0];                      // v_pk_add_f32 candidate pair
    float best0 = c0.x, best1 = c0.y;
    int idx0 = 0, idx1 = 0;
#pragma unroll
    for (int i = 1; i < 32; ++i) {
      float s = lane_bcast(a0, i);
      v2f sv; sv.x = s; sv.y = s;
      v2f cc = sv + tc[i];
      if (cc.x > best0) { best0 = cc.x; idx0 = i; }   // strict >: first-index ties
      if (cc.y > best1) { best1 = cc.y; idx1 = i; }
    }
#pragma unroll
    for (int i = 32; i < 64; ++i) {
      float s = lane_bcast(a1, i - 32);
      v2f sv; sv.x = s; sv.y = s;
      v2f cc = sv + tc[i];
      if (cc.x > best0) { best0 = cc.x; idx0 = i; }
      if (cc.y > best1) { best1 = cc.y; idx1 = i; }
    }

    bp[t * NK + lane]      = (unsigned char)idx0;
    bp[t * NK + lane + 32] = (unsigned char)idx1;

    a0 = best0 + e0;                           // t < L: always valid
    a1 = best1 + e1;
    e0 = ne0; e1 = ne1;
  }

  // Final max/argmax over the 64 frozen alphas: per-lane fold of the two
  // columns, then a 5-stage shuffle butterfly with first-index tie-break.
  float bv = a0; int bi = lane;
  if (a1 > bv) { bv = a1; bi = lane + 32; }
#pragma unroll
  for (int d = 16; d >= 1; d >>= 1) {
    float ov = __shfl_xor(bv, d, 32);
    int   oi = __shfl_xor(bi, d, 32);
    if (ov > bv || (ov == bv && oi < bi)) { bv = ov; bi = oi; }
  }

  // Backtrack: tiny serial chase through LDS by lane 0.
  if (lane == 0) {
    out[(size_t)NB * NT + b] = bv;             // best_score[b]
    int tag = bi;
    unsigned char* tb = (unsigned char*)tagw;
    tb[NT - 1] = (unsigned char)tag;
#pragma unroll 1
    for (int t = NT - 1; t >= 1; --t) {
      int p = bp[t * NK + tag];
      if (t < L) tag = p;                      // where(t < seq_len, bp, tag)
      tb[t - 1] = (unsigned char)tag;
    }
  }
  __syncthreads();

  // Coalesced b128 tag writeback: 4 tags/lane/iteration, int -> float.
  float* __restrict__ ob = out + (size_t)b * NT;
#pragma unroll
  for (int k = 0; k < NT / 128; ++k) {         // 4 iterations
    unsigned int w = tagw[k * 32 + lane];
    float4 f;
    f.x = (float)(w & 0xffu);
    f.y = (float)((w >> 8) & 0xffu);
    f.z = (float)((w >> 16) & 0xffu);
    f.w = (float)(w >> 24);
    *(float4*)(ob + (size_t)(k * 32 + lane) * 4) = f;
  }
}

extern "C" void kernel_launch(void* const* d_in, const int* in_sizes, int n_in,
                              void* d_out, int out_size, void* d_ws, size_t ws_size,
                              hipStream_t stream) {
  const float* pot   = (const float*)d_in[0];  // [2048, 512, 64] f32
  const float* trans = (const float*)d_in[1];  // [64, 64] f32
  const int*   slen  = (const int*)d_in[2];    // [2048, 1] i32
  float* out = (float*)d_out;                  // 2048*512 tags + 2048 scores
  viterbi_fwd_bwd<<<NB, 32, 0, stream>>>(pot, trans, slen, out);
}